// BidirectionalLSTM_89361089561291
// MI455X (gfx1250) — compile-verified
//
#include <hip/hip_runtime.h>
#include <math.h>

// BiLSTM (B=64, T=512, H=512, 2 layers, 2 dirs) for gfx1250.
// cuDNN-style: per (layer,dir) a big parallel WMMA GEMM precomputes
// xp[t][b][g] = x_t @ Wih^T + b (time-major, bias folded); then one small
// per-timestep WMMA kernel does g = xp[t] + h @ Whh^T fused with gates.
// f32 WMMA (V_WMMA_F32_16X16X4_F32) == exact f32 reference math.
// K reduction is split over 4 independent accumulators to break the
// WMMA->WMMA RAW chain (XDL pipelining); xp GEMM register-blocks 16Mx64N.

#define T_SEQ 512
#define BATCH 64
#define HID   512
#define GATES (4 * HID)   // 2048
#define COMBW (2 * HID)   // 1024
#define IN0   256

typedef __attribute__((ext_vector_type(2))) float v2f;
typedef __attribute__((ext_vector_type(8))) float v8f;

__device__ __forceinline__ float sig_(float x) { return 1.0f / (1.0f + __expf(-x)); }

#define WMMA_F32(A, B, C) \
    __builtin_amdgcn_wmma_f32_16x16x4_f32(false, (A), false, (B), (short)0, (C), false, false)

// ---------------------------------------------------------------------------
// Kernel A: xp[(t*B + b)*2048 + n] = sum_k src[b][ts][k] * W[n][k] + bias[n]
//   src: (BATCH, T_SEQ, K) row-major; ts = reverse ? T-1-t : t.
//   One wave = 16M x 64N tile: 1 shared A fragment, 4 B fragments, 4
//   independent accumulators per k-step (good XDL overlap, 5 loads / 4 wmma).
//   Fragments (f32 16x16x4): A lanes0-15 row M=l, K={k,k+1}; lanes16-31 K={k+2,k+3}.
//   B lanes0-15 col N=l, same K split. C/D: VGPR r, lane l -> (M=r+8*(l>>4), N=l&15).
// ---------------------------------------------------------------------------
__global__ __launch_bounds__(256)
void xp_gemm_kernel(const float* __restrict__ src,
                    const float* __restrict__ W,
                    const float* __restrict__ bias,
                    float* __restrict__ out,
                    int K, int reverse)
{
    const int lane  = threadIdx.x & 31;
    const int wave  = threadIdx.x >> 5;
    const int lhalf = lane >> 4;       // 0: K+{0,1}, 1: K+{2,3}
    const int lrow  = lane & 15;

    const int n0 = blockIdx.x * 64;
    const int m0 = (blockIdx.y * 8 + wave) * 16;   // 16-row tile lies in one t (B=64)
    const int t  = m0 / BATCH;
    const int b0 = m0 % BATCH;
    const int ts = reverse ? (T_SEQ - 1 - t) : t;

    const float* arow  = src + ((size_t)(b0 + lrow) * T_SEQ + ts) * K;
    const float* brow0 = W + (size_t)(n0 +  0 + lrow) * K;
    const float* brow1 = W + (size_t)(n0 + 16 + lrow) * K;
    const float* brow2 = W + (size_t)(n0 + 32 + lrow) * K;
    const float* brow3 = W + (size_t)(n0 + 48 + lrow) * K;

    v8f acc0 = {}, acc1 = {}, acc2 = {}, acc3 = {};
#pragma unroll 2
    for (int k = 0; k < K; k += 4) {
        const int ko = k + 2 * lhalf;
        __builtin_prefetch(brow0 + ko + 32, 0, 1);
        v2f a  = *(const v2f*)(arow  + ko);
        v2f b0 = *(const v2f*)(brow0 + ko);
        v2f b1 = *(const v2f*)(brow1 + ko);
        v2f b2 = *(const v2f*)(brow2 + ko);
        v2f b3 = *(const v2f*)(brow3 + ko);
        acc0 = WMMA_F32(a, b0, acc0);
        acc1 = WMMA_F32(a, b1, acc1);
        acc2 = WMMA_F32(a, b2, acc2);
        acc3 = WMMA_F32(a, b3, acc3);
    }

    float* obase = out + ((size_t)t * BATCH + b0) * GATES + n0 + lrow;
    v8f accs[4] = { acc0, acc1, acc2, acc3 };
#pragma unroll
    for (int s = 0; s < 4; ++s) {
        const float bv = bias[n0 + 16 * s + lrow];
#pragma unroll
        for (int r = 0; r < 8; ++r)
            obase[(size_t)(r + 8 * lhalf) * GATES + 16 * s] = accs[s][r] + bv;
    }
}

// ---------------------------------------------------------------------------
// Kernel B: one timestep for both directions of one layer.
//   grid = (HID/16, 2): blockIdx.x -> 16-wide hidden slice j0, blockIdx.y -> dir.
//   512 threads = 16 waves: wave w computes gate (w&3), M-tile (w>>2) of the
//   64x2048 recurrent GEMM restricted to columns gate*512 + [j0, j0+16).
//   K=512 reduction split into 4 accumulator streams (independent WMMAs),
//   xp[t] fragment added at the end. Gate tiles exchanged via LDS; then the
//   c/h update + scatter into the comb output.
// ---------------------------------------------------------------------------
__global__ __launch_bounds__(512)
void lstm_step_kernel(const float* __restrict__ xp_f,
                      const float* __restrict__ xp_b,
                      const float* __restrict__ Whh_f,
                      const float* __restrict__ Whh_b,
                      const float* __restrict__ h_in,   // [dir][BATCH*HID]
                      float* __restrict__ h_out,        // [dir][BATCH*HID]
                      float* __restrict__ c_state,      // [dir][BATCH*HID]
                      float* __restrict__ comb_dst,     // (BATCH, T_SEQ, 2*HID)
                      int t)
{
    __shared__ float gbuf[4][BATCH][17];   // padded to dodge bank conflicts

    const int lane  = threadIdx.x & 31;
    const int wave  = threadIdx.x >> 5;    // 0..15
    const int lhalf = lane >> 4;
    const int lrow  = lane & 15;

    const int dir  = blockIdx.y;
    const int j0   = blockIdx.x * 16;
    const int gate = wave & 3;
    const int m0   = (wave >> 2) * 16;
    const int n0   = gate * HID + j0;

    const float* xp  = dir ? xp_b : xp_f;
    const float* Whh = dir ? Whh_b : Whh_f;
    const float* h   = h_in + (size_t)dir * BATCH * HID;

    const float* arow = h   + (size_t)(m0 + lrow) * HID;
    const float* brow = Whh + (size_t)(n0 + lrow) * HID;

    v8f s0 = {}, s1 = {}, s2 = {}, s3 = {};
#pragma unroll 2
    for (int k = 0; k < HID; k += 16) {
        const int ko = k + 2 * lhalf;
        __builtin_prefetch(brow + ko + 64, 0, 1);
        v2f a0 = *(const v2f*)(arow + ko);
        v2f b0 = *(const v2f*)(brow + ko);
        v2f a1 = *(const v2f*)(arow + ko + 4);
        v2f b1 = *(const v2f*)(brow + ko + 4);
        v2f a2 = *(const v2f*)(arow + ko + 8);
        v2f b2 = *(const v2f*)(brow + ko + 8);
        v2f a3 = *(const v2f*)(arow + ko + 12);
        v2f b3 = *(const v2f*)(brow + ko + 12);
        s0 = WMMA_F32(a0, b0, s0);
        s1 = WMMA_F32(a1, b1, s1);
        s2 = WMMA_F32(a2, b2, s2);
        s3 = WMMA_F32(a3, b3, s3);
    }

    // acc = xp tile + sum of the 4 partial accumulators.
    const float* xrow = xp + ((size_t)t * BATCH + m0) * GATES + n0 + lrow;
#pragma unroll
    for (int r = 0; r < 8; ++r) {
        const float g = xrow[(size_t)(r + 8 * lhalf) * GATES]
                      + ((s0[r] + s1[r]) + (s2[r] + s3[r]));
        gbuf[gate][m0 + r + 8 * lhalf][lrow] = g;
    }
    __syncthreads();

    // Gate fusion for the (BATCH x 16) hidden slice owned by this block.
    float* cs = c_state + (size_t)dir * BATCH * HID;
    float* ho = h_out   + (size_t)dir * BATCH * HID;
    const int t_out = dir ? (T_SEQ - 1 - t) : t;
    const int coff  = dir ? HID : 0;
    for (int idx = threadIdx.x; idx < BATCH * 16; idx += 512) {
        const int b  = idx >> 4;
        const int j  = idx & 15;
        const int jj = j0 + j;
        const float ig = sig_(gbuf[0][b][j]);
        const float fg = sig_(gbuf[1][b][j]);
        const float gg = tanhf(gbuf[2][b][j]);
        const float og = sig_(gbuf[3][b][j]);
        const float cn = fg * cs[(size_t)b * HID + jj] + ig * gg;
        const float hn = og * tanhf(cn);
        cs[(size_t)b * HID + jj] = cn;
        ho[(size_t)b * HID + jj] = hn;
        comb_dst[((size_t)b * T_SEQ + t_out) * COMBW + coff + jj] = hn;
    }
}

// ---------------------------------------------------------------------------
// State init: h0/c0 layout (4, B, H) = [fwd0, fwd1, bwd0, bwd1].
// Reference quirk: bwd layer1 uses c_0[1] (not c_0[3]).
// ---------------------------------------------------------------------------
__global__ void init_state_kernel(const float* __restrict__ h0,
                                  const float* __restrict__ c0,
                                  int layer,
                                  float* __restrict__ h_state0,
                                  float* __restrict__ c_state)
{
    const int i = blockIdx.x * blockDim.x + threadIdx.x;   // [0, 2*B*H)
    const int dir = i / (BATCH * HID);
    const int off = i % (BATCH * HID);
    const int hidx = dir * 2 + layer;
    const int cidx = (layer == 1 && dir == 1) ? 1 : hidx;
    h_state0[i] = h0[(size_t)hidx * BATCH * HID + off];
    c_state[i]  = c0[(size_t)cidx * BATCH * HID + off];
}

// Final states -> tail of d_out in order [h_f, h_b, c_f, c_b].
__global__ void write_finals_kernel(const float* __restrict__ h_fin,
                                    const float* __restrict__ c_state,
                                    float* __restrict__ out_tail)
{
    const int i = blockIdx.x * blockDim.x + threadIdx.x;   // [0, 4*B*H)
    const int which = i / (BATCH * HID);
    const int off   = i % (BATCH * HID);
    out_tail[i] = (which < 2) ? h_fin[(size_t)which * BATCH * HID + off]
                              : c_state[(size_t)(which - 2) * BATCH * HID + off];
}

// ---------------------------------------------------------------------------
extern "C" void kernel_launch(void* const* d_in, const int* in_sizes, int n_in,
                              void* d_out, int out_size, void* d_ws, size_t ws_size,
                              hipStream_t stream)
{
    (void)n_in; (void)out_size; (void)ws_size;

    const float* x  = (const float*)d_in[0];   // (B, T, 256)
    const float* h0 = (const float*)d_in[1];   // (4, B, H)
    const float* c0 = (const float*)d_in[2];   // (4, B, H)

    // Params pytree order detection via sizes:
    //   sorted-key flatten (jax.tree): bwd0{Whh,Wih,b}, bwd1, fwd0, fwd1
    //   insertion order:               fwd0{Wih,Whh,b}, fwd1, bwd0, bwd1
    const float *Wih[2][2], *Whh[2][2], *Bv[2][2];   // [dir][layer], dir0 = fwd
    const bool sorted = (in_sizes[3] == GATES * HID);
    if (sorted) {
        const int base[2][2] = { {9, 12}, {3, 6} };  // [dir][layer]
        for (int d = 0; d < 2; ++d)
            for (int l = 0; l < 2; ++l) {
                Whh[d][l] = (const float*)d_in[base[d][l] + 0];
                Wih[d][l] = (const float*)d_in[base[d][l] + 1];
                Bv [d][l] = (const float*)d_in[base[d][l] + 2];
            }
    } else {
        const int base[2][2] = { {3, 6}, {9, 12} };
        for (int d = 0; d < 2; ++d)
            for (int l = 0; l < 2; ++l) {
                Wih[d][l] = (const float*)d_in[base[d][l] + 0];
                Whh[d][l] = (const float*)d_in[base[d][l] + 1];
                Bv [d][l] = (const float*)d_in[base[d][l] + 2];
            }
    }

    // Workspace layout (floats):
    //   comb0: B*T*2H, xp_f/xp_b: T*B*4H each, h_state: 2 parity * 2 dir * B*H,
    //   c_state: 2 dir * B*H. Total ~169M floats.
    float* ws    = (float*)d_ws;
    float* comb0 = ws;
    float* xpf   = comb0 + (size_t)BATCH * T_SEQ * COMBW;
    float* xpb   = xpf   + (size_t)T_SEQ * BATCH * GATES;
    float* hst   = xpb   + (size_t)T_SEQ * BATCH * GATES;
    float* cst   = hst   + (size_t)4 * BATCH * HID;
    float* outF  = (float*)d_out;

    const dim3 ggrid(GATES / 64, (T_SEQ * BATCH) / (16 * 8));   // (32, 256)

    for (int layer = 0; layer < 2; ++layer) {
        const float* src = layer ? comb0 : x;
        const int K = layer ? COMBW : IN0;

        xp_gemm_kernel<<<ggrid, 256, 0, stream>>>(src, Wih[0][layer], Bv[0][layer], xpf, K, 0);
        xp_gemm_kernel<<<ggrid, 256, 0, stream>>>(src, Wih[1][layer], Bv[1][layer], xpb, K, 1);

        init_state_kernel<<<(2 * BATCH * HID) / 256, 256, 0, stream>>>(h0, c0, layer, hst, cst);

        float* comb_dst = layer ? outF : comb0;
        for (int t = 0; t < T_SEQ; ++t) {
            const float* hin = hst + (size_t)(t & 1) * 2 * BATCH * HID;
            float* hout      = hst + (size_t)((t + 1) & 1) * 2 * BATCH * HID;
            lstm_step_kernel<<<dim3(HID / 16, 2), 512, 0, stream>>>(
                xpf, xpb, Whh[0][layer], Whh[1][layer], hin, hout, cst, comb_dst, t);
        }
    }

    // T_SEQ is even -> final h lives in parity 0 (== hst).
    write_finals_kernel<<<(4 * BATCH * HID) / 256, 256, 0, stream>>>(
        hst, cst, outF + (size_t)BATCH * T_SEQ * COMBW);
}